// Echo_72739566125444
// MI455X (gfx1250) — compile-verified
//
#include <hip/hip_runtime.h>
#include <cstdint>

// Problem constants (from the reference)
constexpr int   B_   = 64;
constexpr int   T_   = 480000;
constexpr int   K_   = 4800;
constexpr int   L_   = T_ - K_ + 1;   // 475201 valid conv outputs
constexpr float VOL  = 0.3f;

constexpr int TPB    = 256;           // 8 waves (wave32)
constexpr int TILE   = TPB * 4;       // 1024 floats per block-tile
constexpr int TILES  = (T_ + TILE - 1) / TILE;  // 469
constexpr int GX     = 32;            // reduce blocks per row (grid-stride over tiles)
constexpr int NPART  = GX * B_;       // 2048 partials
constexpr int STAGE_N = TILE + 8;     // need 1028 staged floats; pad

typedef float vf4 __attribute__((ext_vector_type(4)));  // native vector for NT stores

// ---- CDNA5 async global->LDS staging -------------------------------------
#if defined(__AMDGCN__) && __has_builtin(__builtin_amdgcn_global_load_async_to_lds_b128)
#define ECHO_ASYNC 1
typedef int v4i __attribute__((ext_vector_type(4)));
typedef __attribute__((address_space(1))) v4i as1_v4i;   // global v4i
typedef __attribute__((address_space(3))) v4i as3_v4i;   // LDS v4i
#else
#define ECHO_ASYNC 0
#endif

__device__ __forceinline__ void stage_copy16(const float* gsrc, float* ldst) {
#if ECHO_ASYNC
    // fire-and-forget 16B global -> LDS DMA, tracked by ASYNCcnt
    __builtin_amdgcn_global_load_async_to_lds_b128(
        (as1_v4i*)(unsigned long long)(uintptr_t)gsrc,
        (as3_v4i*)(unsigned int)(uintptr_t)ldst,
        0 /*imm offset*/, 0 /*cpol*/);
#else
    *(float4*)ldst = *(const float4*)gsrc;
#endif
}

__device__ __forceinline__ void stage_wait() {
#if ECHO_ASYNC
#if __has_builtin(__builtin_amdgcn_s_wait_asynccnt)
    __builtin_amdgcn_s_wait_asynccnt(0);
#else
    asm volatile("s_wait_asynccnt 0" ::: "memory");
#endif
#endif
}

// Stage echo window [t0+K-4, t0+K-4+1027] of `row` into stage[0..1027].
// Chunk start addresses are 16B aligned (t0, K-4=4796 both multiples of 4).
// Chunks that would start at >= T are clamped to T-4; their data is only ever
// consumed for n >= L, where the output is forced to zero anyway.
__device__ __forceinline__ void stage_echo(const float* row, float* stage, int t0) {
    const int tid = threadIdx.x;
    const int s0  = t0 + (K_ - 4);                 // 4796 offset
    int a  = s0 + tid * 4;
    int ga = (a <= T_ - 4) ? a : (T_ - 4);
    stage_copy16(row + ga, stage + tid * 4);
    if (tid == 0) {
        int a2  = s0 + TILE;
        int ga2 = (a2 <= T_ - 4) ? a2 : (T_ - 4);
        stage_copy16(row + ga2, stage + TILE);     // stage[1024..1027]
    }
    stage_wait();        // my wave's async copies have landed in LDS
    __syncthreads();     // all waves' copies visible
}

// ---- block-wide max reduction (wave32 shuffles + LDS) --------------------
__device__ __forceinline__ void blockMax2(float& mX, float& mR,
                                          float* smx, float* smr) {
    #pragma unroll
    for (int off = 16; off > 0; off >>= 1) {
        mX = fmaxf(mX, __shfl_xor(mX, off, 32));
        mR = fmaxf(mR, __shfl_xor(mR, off, 32));
    }
    const int wid = threadIdx.x >> 5;
    if ((threadIdx.x & 31) == 0) { smx[wid] = mX; smr[wid] = mR; }
    __syncthreads();
    if (threadIdx.x == 0) {
        #pragma unroll
        for (int i = 1; i < TPB / 32; ++i) {
            mX = fmaxf(mX, smx[i]);
            mR = fmaxf(mR, smr[i]);
        }
    }
}

// ---- pass 1: per-block partial max|x| and max|x[n]+V*x[n+K-1]| -----------
__global__ void echoReduceKernel(const float* __restrict__ x,
                                 float* __restrict__ px,
                                 float* __restrict__ pr) {
    __shared__ float stage[STAGE_N];
    __shared__ float smx[TPB / 32], smr[TPB / 32];

    const int    b   = blockIdx.y;
    const float* row = x + (size_t)b * T_;
    const int    tid = threadIdx.x;

    float mX = 0.f, mR = 0.f;

    for (int tile = blockIdx.x; tile < TILES; tile += GX) {
        const int t0  = tile * TILE;
        stage_echo(row, stage, t0);

        const int idx = t0 + tid * 4;
        const int gi  = (idx <= T_ - 4) ? idx : (T_ - 4);
        const float4 m = *(const float4*)(row + gi);   // aligned b128

        // prefetch this thread's next tile (global_prefetch_b8)
        const int nidx = idx + GX * TILE;
        if (nidx <= T_ - 4) __builtin_prefetch(row + nidx, 0, 0);

        mX = fmaxf(mX, fmaxf(fmaxf(fabsf(m.x), fabsf(m.y)),
                             fmaxf(fabsf(m.z), fabsf(m.w))));

        const float mv[4] = {m.x, m.y, m.z, m.w};
        const float* e = &stage[tid * 4 + 3];   // stage[s] = row[t0+4796+s]
        #pragma unroll
        for (int j = 0; j < 4; ++j) {
            const int n = idx + j;
            if (n < L_) {
                const float y = mv[j] + VOL * e[j];   // e[j] = row[n+K-1]
                mR = fmaxf(mR, fabsf(y));
            }
        }
        __syncthreads();   // done reading stage before next iteration restages
    }

    blockMax2(mX, mR, smx, smr);
    if (tid == 0) {
        const int p = blockIdx.y * GX + blockIdx.x;
        px[p] = mX;
        pr[p] = mR;
    }
}

// ---- pass 2: fold 2048 partials into ws[0]=max|x|, ws[1]=max|y| ----------
__global__ void echoFinalReduceKernel(float* __restrict__ ws) {
    __shared__ float smx[TPB / 32], smr[TPB / 32];
    const float* px = ws + 2;
    const float* pr = ws + 2 + NPART;
    float mX = 0.f, mR = 0.f;
    for (int i = threadIdx.x; i < NPART; i += TPB) {
        mX = fmaxf(mX, px[i]);
        mR = fmaxf(mR, pr[i]);
    }
    blockMax2(mX, mR, smx, smr);
    if (threadIdx.x == 0) {
        ws[0] = mX;
        ws[1] = (mR > 0.f) ? mR : 1.f;   // guard degenerate all-zero input
    }
}

// ---- pass 3: y[n] = (x[n] + V*x[n+K-1]) * (max|x|/max|y|), zero-padded ---
__global__ void echoApplyKernel(const float* __restrict__ x,
                                float* __restrict__ out,
                                const float* __restrict__ ws) {
    __shared__ float stage[STAGE_N];
    const float g = ws[0] / ws[1];

    const int    b   = blockIdx.y;
    const float* row = x + (size_t)b * T_;
    const int    tid = threadIdx.x;
    const int    t0  = blockIdx.x * TILE;

    stage_echo(row, stage, t0);

    const int idx = t0 + tid * 4;
    const int gi  = (idx <= T_ - 4) ? idx : (T_ - 4);
    const float4 m = *(const float4*)(row + gi);

    const float mv[4] = {m.x, m.y, m.z, m.w};
    const float* e = &stage[tid * 4 + 3];
    float r[4];
    #pragma unroll
    for (int j = 0; j < 4; ++j) {
        const int n = idx + j;
        r[j] = (n < L_) ? (mv[j] + VOL * e[j]) * g : 0.f;
    }

    float* orow = out + (size_t)b * T_;
    if (idx + 3 < T_) {
        // non-temporal: keep the 123 MB write stream from evicting x out of L2
        const vf4 v = {r[0], r[1], r[2], r[3]};
        __builtin_nontemporal_store(v, (vf4*)(orow + idx));
    } else {
        #pragma unroll
        for (int j = 0; j < 4; ++j)
            if (idx + j < T_) orow[idx + j] = r[j];
    }
}

// ---- mask passthrough: bool bytes -> float -------------------------------
__global__ void maskCopyKernel(const unsigned char* __restrict__ m,
                               float* __restrict__ out, int n4) {
    const int stride = gridDim.x * blockDim.x;
    for (int i = blockIdx.x * blockDim.x + threadIdx.x; i < n4; i += stride) {
        const uchar4 v = ((const uchar4*)m)[i];
        const vf4 f = {v.x ? 1.f : 0.f, v.y ? 1.f : 0.f,
                       v.z ? 1.f : 0.f, v.w ? 1.f : 0.f};
        __builtin_nontemporal_store(f, (vf4*)out + i);
    }
}

extern "C" void kernel_launch(void* const* d_in, const int* in_sizes, int n_in,
                              void* d_out, int out_size, void* d_ws, size_t ws_size,
                              hipStream_t stream) {
    const float*         x    = (const float*)d_in[0];
    const unsigned char* mask = (n_in > 1) ? (const unsigned char*)d_in[1] : nullptr;
    float* out = (float*)d_out;
    float* ws  = (float*)d_ws;   // [0]=max|x| [1]=max|y| [2..]=partials (16.4 KB)

    const dim3 blk(TPB);
    echoReduceKernel<<<dim3(GX, B_), blk, 0, stream>>>(x, ws + 2, ws + 2 + NPART);
    echoFinalReduceKernel<<<dim3(1), blk, 0, stream>>>(ws);
    echoApplyKernel<<<dim3(TILES, B_), blk, 0, stream>>>(x, out, ws);

    const long long N = (long long)B_ * T_;
    if (mask && (long long)out_size >= 2 * N) {
        maskCopyKernel<<<dim3(1024), blk, 0, stream>>>(mask, out + N, (int)(N / 4));
    }
}